// XLNetMainLayer_20624432955784
// MI455X (gfx1250) — compile-verified
//
#include <hip/hip_runtime.h>
#include <hip/hip_bf16.h>
#include <math.h>

// XLNet main layer forward on gfx1250 (CDNA5), fp32 WMMA 16x16x4 path.
// L=12, D=768, H=12, DH=64, DI=3072, S=512, B=2.

#define D_MODEL 768
#define N_HEAD  12
#define D_HEAD  64
#define D_FF    3072
#define SEQ     512
#define NBATCH  2
#define NTOK    1024   // SEQ*NBATCH
#define RLEN    1024   // 2*SEQ
#define NLAYER  12
#define LN_EPS  1e-3f

typedef __attribute__((ext_vector_type(2))) float v2f;
typedef __attribute__((ext_vector_type(4))) float v4f;
typedef __attribute__((ext_vector_type(8))) float v8f;

__device__ __forceinline__ float gelu_f(float x) {
  float x3 = x * x * x;
  return 0.5f * x * (1.0f + tanhf(0.7978845608028654f * (x + 0.044715f * x3)));
}

// ---------------------------------------------------------------------------
// Generic fp32 WMMA GEMM:  C = A * B  (or A * B^T when BT).
// Block: 128 threads (4 waves). Block tile 64x64, K chunk 16.
// Wave w computes rows [w*16, w*16+16) x 64 cols as 4 accumulators of 16x16.
// Batched via blockIdx.z: per-operand offset = Outer*(z/inner) + Inner*(z%inner).
// Requires: M,N multiples of 64; K multiple of 16; all leading dims mult of 4.
// OP: 0 = none, 1 = gelu (applied after optional bias).
// ---------------------------------------------------------------------------
template<bool BT, int OP>
__global__ __launch_bounds__(128) void gemm_wmma_f32(
    const float* __restrict__ A, const float* __restrict__ Bm,
    float* __restrict__ C, const float* __restrict__ bias,
    int M, int N, int K, int lda, int ldb, int ldc,
    long long aOuter, long long aInner,
    long long bOuter, long long bInner,
    long long cOuter, long long cInner, int inner)
{
  (void)M; (void)N;
  __shared__ float As[64][20];   // row stride 80B -> 16B-aligned cols {0,4,8,12}
  __shared__ float Bs[16][68];   // row stride 272B -> 16B-aligned

  const int z  = blockIdx.z;
  const int zo = z / inner, zi = z - zo * inner;
  A  += aOuter * zo + aInner * zi;
  Bm += bOuter * zo + bInner * zi;
  C  += cOuter * zo + cInner * zi;

  const int tid  = threadIdx.x;
  const int wave = tid >> 5;
  const int lane = tid & 31;
  const int hf   = lane >> 4;   // lane half (0: lanes 0-15, 1: lanes 16-31)
  const int lr   = lane & 15;
  const int m0 = blockIdx.y * 64;
  const int n0 = blockIdx.x * 64;
  const int mw = wave * 16;

  v8f acc[4] = {};

  for (int kc = 0; kc < K; kc += 16) {
    // ---- stage A tile: 64 rows x 16 K ----
    #pragma unroll
    for (int h2 = 0; h2 < 2; ++h2) {
      int e = tid * 4 + h2 * 512;
      int r = e >> 4, c = e & 15;
      v4f a4 = *(const v4f*)(A + (long long)(m0 + r) * lda + kc + c);
      As[r][c + 0] = a4.x; As[r][c + 1] = a4.y;
      As[r][c + 2] = a4.z; As[r][c + 3] = a4.w;
    }
    // ---- stage B tile: 16 K x 64 cols ----
    if (BT) {
      #pragma unroll
      for (int h2 = 0; h2 < 2; ++h2) {
        int e = tid * 4 + h2 * 512;
        int nn = e >> 4, kk = e & 15;
        v4f b4 = *(const v4f*)(Bm + (long long)(n0 + nn) * ldb + kc + kk);
        Bs[kk + 0][nn] = b4.x; Bs[kk + 1][nn] = b4.y;
        Bs[kk + 2][nn] = b4.z; Bs[kk + 3][nn] = b4.w;
      }
    } else {
      #pragma unroll
      for (int h2 = 0; h2 < 2; ++h2) {
        int e = tid * 4 + h2 * 512;
        int kk = e >> 6, nn = e & 63;
        v4f b4 = *(const v4f*)(Bm + (long long)(kc + kk) * ldb + n0 + nn);
        Bs[kk][nn + 0] = b4.x; Bs[kk][nn + 1] = b4.y;
        Bs[kk][nn + 2] = b4.z; Bs[kk][nn + 3] = b4.w;
      }
    }
    __syncthreads();

    // ---- compute: 4 K-steps of 4, 4 col tiles each ----
    #pragma unroll
    for (int kk = 0; kk < 16; kk += 4) {
      v2f afrag;   // A 16x4 layout: lanes 0-15 K={kk,kk+1}; lanes 16-31 K={kk+2,kk+3}
      afrag.x = As[mw + lr][kk + 2 * hf + 0];
      afrag.y = As[mw + lr][kk + 2 * hf + 1];
      #pragma unroll
      for (int ct = 0; ct < 4; ++ct) {
        v2f bfrag;  // B 4x16 layout mirrored
        bfrag.x = Bs[kk + 2 * hf + 0][ct * 16 + lr];
        bfrag.y = Bs[kk + 2 * hf + 1][ct * 16 + lr];
        acc[ct] = __builtin_amdgcn_wmma_f32_16x16x4_f32(
            false, afrag, false, bfrag, (short)0, acc[ct], false, false);
      }
    }
    __syncthreads();
  }

  // ---- store: C 16x16 layout: VGPR v -> row v + 8*hf, col lr ----
  #pragma unroll
  for (int ct = 0; ct < 4; ++ct) {
    int col = n0 + ct * 16 + lr;
    #pragma unroll
    for (int vv = 0; vv < 8; ++vv) {
      int row = m0 + mw + vv + hf * 8;
      float val = acc[ct][vv];
      if (bias) val += bias[col];
      if (OP == 1) val = gelu_f(val);
      C[(long long)row * ldc + col] = val;
    }
  }
}

// ---------------------------------------------------------------------------
// Elementwise / reduction kernels
// ---------------------------------------------------------------------------
__global__ __launch_bounds__(256) void embed_kernel(
    const int* __restrict__ inp_k, const float* __restrict__ lut,
    float* __restrict__ h)
{
  int t = blockIdx.x;
  long long tok = inp_k[t];
  int tid = threadIdx.x;
  #pragma unroll
  for (int u = 0; u < 3; ++u)
    h[(long long)t * D_MODEL + tid + u * 256] = lut[tok * D_MODEL + tid + u * 256];
}

__global__ __launch_bounds__(256) void posemb_kernel(float* __restrict__ r2)
{
  int p = blockIdx.x;                       // 0..1023
  float pos = (float)(SEQ - p);             // 512 .. -511
  #pragma unroll
  for (int u = 0; u < 3; ++u) {
    int d = threadIdx.x + u * 256;
    int j = (d < 384) ? d : d - 384;
    // inv_freq = 10000^(-2j/768)
    float freq = expf(-(float)(2 * j) * (1.0f / 768.0f) * 9.210340371976184f);
    float ang = pos * freq;
    r2[(long long)p * D_MODEL + d] = (d < 384) ? sinf(ang) : cosf(ang);
  }
}

__global__ __launch_bounds__(256) void add_rowbias_kernel(
    const float* __restrict__ q, const float* __restrict__ bias,
    float* __restrict__ out)
{
  int t = blockIdx.x, tid = threadIdx.x;
  #pragma unroll
  for (int u = 0; u < 3; ++u) {
    int d = tid + u * 256;
    out[(long long)t * D_MODEL + d] = q[(long long)t * D_MODEL + d] + bias[d];
  }
}

// ef[t][n][s] = sum_d qs[t][n*64+d] * seg_embed[l][s][n][d]
__global__ __launch_bounds__(32) void ef_kernel(
    const float* __restrict__ qs, const float* __restrict__ segw,
    float* __restrict__ ef)
{
  int t = blockIdx.x;
  int tid = threadIdx.x;
  if (tid < 2 * N_HEAD) {
    int n = tid >> 1, s = tid & 1;
    const float* qp = qs + (long long)t * D_MODEL + n * D_HEAD;
    const float* wp = segw + (long long)(s * N_HEAD + n) * D_HEAD;
    float acc = 0.0f;
    #pragma unroll 8
    for (int d = 0; d < D_HEAD; ++d) acc += qp[d] * wp[d];
    ef[((long long)t * N_HEAD + n) * 2 + s] = acc;
  }
}

// One wave per score row (b,n,i): fuse rel_shift + segment select + mask,
// softmax over j (512 keys), write prob in place of ac.
__global__ __launch_bounds__(256) void softmax_kernel(
    float* __restrict__ ac, const float* __restrict__ bd,
    const float* __restrict__ ef, const int* __restrict__ seg_ids,
    const int* __restrict__ input_mask)
{
  int wave = threadIdx.x >> 5, lane = threadIdx.x & 31;
  int row = blockIdx.x * 8 + wave;            // 0..(2*12*512-1)
  int i  = row & (SEQ - 1);
  int bn = row >> 9;                          // b*N_HEAD + n
  int b = bn / N_HEAD, n = bn - b * N_HEAD;

  float* acrow = ac + (long long)row * SEQ;
  const float* bdrow = bd + (long long)row * RLEN;
  const float* efrow = ef + ((long long)(i * NBATCH + b) * N_HEAD + n) * 2;
  int segi = seg_ids[i * NBATCH + b];
  const float scale = 0.125f;                 // DH^-0.5

  float sv[16];
  float mx = -3.0e38f;
  #pragma unroll
  for (int it = 0; it < 16; ++it) {
    int j = lane + it * 32;
    float s = acrow[j] + bdrow[j + SEQ - i];  // rel_shift remap
    int sd = (seg_ids[j * NBATCH + b] != segi) ? 1 : 0;
    s = (s + efrow[sd]) * scale;
    s -= 1.0e30f * (float)input_mask[j * NBATCH + b];
    sv[it] = s;
    mx = fmaxf(mx, s);
  }
  #pragma unroll
  for (int o = 16; o > 0; o >>= 1) mx = fmaxf(mx, __shfl_xor(mx, o, 32));
  float sum = 0.0f;
  #pragma unroll
  for (int it = 0; it < 16; ++it) { sv[it] = expf(sv[it] - mx); sum += sv[it]; }
  #pragma unroll
  for (int o = 16; o > 0; o >>= 1) sum += __shfl_xor(sum, o, 32);
  float inv = 1.0f / sum;
  #pragma unroll
  for (int it = 0; it < 16; ++it) acrow[lane + it * 32] = sv[it] * inv;
}

// h = LayerNorm(x + h) * g + b  (in place over h), one block per token
__global__ __launch_bounds__(256) void add_ln_kernel(
    const float* __restrict__ x, float* __restrict__ h,
    const float* __restrict__ g, const float* __restrict__ bb)
{
  int t = blockIdx.x, tid = threadIdx.x;
  __shared__ float red[256];
  long long base = (long long)t * D_MODEL;
  float v0 = x[base + tid]       + h[base + tid];
  float v1 = x[base + tid + 256] + h[base + tid + 256];
  float v2 = x[base + tid + 512] + h[base + tid + 512];
  red[tid] = v0 + v1 + v2;
  __syncthreads();
  for (int o = 128; o > 0; o >>= 1) { if (tid < o) red[tid] += red[tid + o]; __syncthreads(); }
  float mu = red[0] * (1.0f / (float)D_MODEL);
  __syncthreads();
  float d0 = v0 - mu, d1 = v1 - mu, d2 = v2 - mu;
  red[tid] = d0 * d0 + d1 * d1 + d2 * d2;
  __syncthreads();
  for (int o = 128; o > 0; o >>= 1) { if (tid < o) red[tid] += red[tid + o]; __syncthreads(); }
  float rstd = rsqrtf(red[0] * (1.0f / (float)D_MODEL) + LN_EPS);
  h[base + tid]       = d0 * rstd * g[tid]       + bb[tid];
  h[base + tid + 256] = d1 * rstd * g[tid + 256] + bb[tid + 256];
  h[base + tid + 512] = d2 * rstd * g[tid + 512] + bb[tid + 512];
}

// ---------------------------------------------------------------------------
extern "C" void kernel_launch(void* const* d_in, const int* in_sizes, int n_in,
                              void* d_out, int out_size, void* d_ws, size_t ws_size,
                              hipStream_t stream)
{
  (void)in_sizes; (void)n_in; (void)out_size; (void)ws_size;
  const int*   inp_k      = (const int*)  d_in[0];
  const int*   seg_ids    = (const int*)  d_in[1];
  const int*   input_mask = (const int*)  d_in[2];
  const float* lut        = (const float*)d_in[3];
  const float* Wq         = (const float*)d_in[4];
  const float* Wk         = (const float*)d_in[5];
  const float* Wv         = (const float*)d_in[6];
  const float* Wr         = (const float*)d_in[7];
  const float* Wo         = (const float*)d_in[8];
  const float* r_w_bias   = (const float*)d_in[9];
  const float* r_r_bias   = (const float*)d_in[10];
  const float* r_s_bias   = (const float*)d_in[11];
  const float* seg_embed  = (const float*)d_in[12];
  const float* ln_attn_g  = (const float*)d_in[13];
  const float* ln_attn_b  = (const float*)d_in[14];
  const float* ln_ff_g    = (const float*)d_in[15];
  const float* ln_ff_b    = (const float*)d_in[16];
  const float* ff_w1      = (const float*)d_in[17];
  const float* ff_b1      = (const float*)d_in[18];
  const float* ff_w2      = (const float*)d_in[19];
  const float* ff_b2      = (const float*)d_in[20];

  const long long TD = (long long)NTOK * D_MODEL;   // 786432
  float* ws = (float*)d_ws;
  float* h    = ws; ws += TD;
  float* r2   = ws; ws += TD;
  float* q    = ws; ws += TD;
  float* kbuf = ws; ws += TD;
  float* vbuf = ws; ws += TD;
  float* kr   = ws; ws += TD;
  float* qb   = ws; ws += TD;
  float* vec  = ws; ws += TD;
  float* attn = ws; ws += TD;
  float* ff1b = ws; ws += (long long)NTOK * D_FF;
  float* ff2b = ws; ws += TD;
  float* efb  = ws; ws += (long long)NTOK * N_HEAD * 2;
  float* ac   = ws; ws += (long long)NBATCH * N_HEAD * SEQ * SEQ;   // also prob
  float* bdb  = ws; ws += (long long)NBATCH * N_HEAD * SEQ * RLEN;

  embed_kernel<<<NTOK, 256, 0, stream>>>(inp_k, lut, h);
  posemb_kernel<<<RLEN, 256, 0, stream>>>(r2);

  const dim3 gProj(D_MODEL / 64, NTOK / 64, 1);   // 12 x 16
  const dim3 gAC(SEQ / 64, SEQ / 64, NBATCH * N_HEAD);
  const dim3 gBD(RLEN / 64, SEQ / 64, NBATCH * N_HEAD);
  const dim3 gVec(D_HEAD / 64, SEQ / 64, NBATCH * N_HEAD);
  const dim3 gFF1(D_FF / 64, NTOK / 64, 1);

  for (int l = 0; l < NLAYER; ++l) {
    const float* wq = Wq + (long long)l * D_MODEL * D_MODEL;
    const float* wk = Wk + (long long)l * D_MODEL * D_MODEL;
    const float* wv = Wv + (long long)l * D_MODEL * D_MODEL;
    const float* wr = Wr + (long long)l * D_MODEL * D_MODEL;
    const float* wo = Wo + (long long)l * D_MODEL * D_MODEL;
    const float* bw = r_w_bias + (long long)l * D_MODEL;
    const float* br = r_r_bias + (long long)l * D_MODEL;
    const float* bs = r_s_bias + (long long)l * D_MODEL;
    const float* sw = seg_embed + (long long)l * 2 * D_MODEL;
    const float* w1 = ff_w1 + (long long)l * D_MODEL * D_FF;
    const float* b1 = ff_b1 + (long long)l * D_FF;
    const float* w2 = ff_w2 + (long long)l * D_FF * D_MODEL;
    const float* b2 = ff_b2 + (long long)l * D_MODEL;

    // projections: q/k/v from h, kr from pos-emb (batch-collapsed)
    gemm_wmma_f32<false,0><<<gProj, 128, 0, stream>>>(h,  wq, q,    nullptr,
        NTOK, D_MODEL, D_MODEL, D_MODEL, D_MODEL, D_MODEL, 0,0, 0,0, 0,0, 1);
    gemm_wmma_f32<false,0><<<gProj, 128, 0, stream>>>(h,  wk, kbuf, nullptr,
        NTOK, D_MODEL, D_MODEL, D_MODEL, D_MODEL, D_MODEL, 0,0, 0,0, 0,0, 1);
    gemm_wmma_f32<false,0><<<gProj, 128, 0, stream>>>(h,  wv, vbuf, nullptr,
        NTOK, D_MODEL, D_MODEL, D_MODEL, D_MODEL, D_MODEL, 0,0, 0,0, 0,0, 1);
    gemm_wmma_f32<false,0><<<gProj, 128, 0, stream>>>(r2, wr, kr,   nullptr,
        RLEN, D_MODEL, D_MODEL, D_MODEL, D_MODEL, D_MODEL, 0,0, 0,0, 0,0, 1);

    // ac = (q + r_w_bias) . k^T, per (b, head)
    add_rowbias_kernel<<<NTOK, 256, 0, stream>>>(q, bw, qb);
    gemm_wmma_f32<true,0><<<gAC, 128, 0, stream>>>(qb, kbuf, ac, nullptr,
        SEQ, SEQ, D_HEAD, NBATCH * D_MODEL, NBATCH * D_MODEL, SEQ,
        (long long)D_MODEL, (long long)D_HEAD,
        (long long)D_MODEL, (long long)D_HEAD,
        (long long)N_HEAD * SEQ * SEQ, (long long)SEQ * SEQ, N_HEAD);

    // bd_raw = (q + r_r_bias) . kr^T  (kr is batch-invariant)
    add_rowbias_kernel<<<NTOK, 256, 0, stream>>>(q, br, qb);
    gemm_wmma_f32<true,0><<<gBD, 128, 0, stream>>>(qb, kr, bdb, nullptr,
        SEQ, RLEN, D_HEAD, NBATCH * D_MODEL, D_MODEL, RLEN,
        (long long)D_MODEL, (long long)D_HEAD,
        0LL, (long long)D_HEAD,
        (long long)N_HEAD * SEQ * RLEN, (long long)SEQ * RLEN, N_HEAD);

    // ef[t][n][s] = (q + r_s_bias) . seg_embed
    add_rowbias_kernel<<<NTOK, 256, 0, stream>>>(q, bs, qb);
    ef_kernel<<<NTOK, 32, 0, stream>>>(qb, sw, efb);

    // softmax (ac + shifted bd + ef)*scale - mask -> prob (in place in ac)
    softmax_kernel<<<NBATCH * N_HEAD * SEQ / 8, 256, 0, stream>>>(
        ac, bdb, efb, seg_ids, input_mask);

    // vec = prob . v, per (b, head)
    gemm_wmma_f32<false,0><<<gVec, 128, 0, stream>>>(ac, vbuf, vec, nullptr,
        SEQ, D_HEAD, SEQ, SEQ, NBATCH * D_MODEL, NBATCH * D_MODEL,
        (long long)N_HEAD * SEQ * SEQ, (long long)SEQ * SEQ,
        (long long)D_MODEL, (long long)D_HEAD,
        (long long)D_MODEL, (long long)D_HEAD, N_HEAD);

    // attn_out = vec . Wo^T ;  h = LN(attn_out + h)
    gemm_wmma_f32<true,0><<<gProj, 128, 0, stream>>>(vec, wo, attn, nullptr,
        NTOK, D_MODEL, D_MODEL, D_MODEL, D_MODEL, D_MODEL, 0,0, 0,0, 0,0, 1);
    add_ln_kernel<<<NTOK, 256, 0, stream>>>(attn, h,
        ln_attn_g + (long long)l * D_MODEL, ln_attn_b + (long long)l * D_MODEL);

    // FFN: gelu(h.W1 + b1).W2 + b2 ;  h = LN(ffn + h)
    gemm_wmma_f32<false,1><<<gFF1, 128, 0, stream>>>(h, w1, ff1b, b1,
        NTOK, D_FF, D_MODEL, D_MODEL, D_FF, D_FF, 0,0, 0,0, 0,0, 1);
    gemm_wmma_f32<false,0><<<gProj, 128, 0, stream>>>(ff1b, w2, ff2b, b2,
        NTOK, D_MODEL, D_FF, D_FF, D_MODEL, D_MODEL, 0,0, 0,0, 0,0, 1);
    add_ln_kernel<<<NTOK, 256, 0, stream>>>(ff2b, h,
        ln_ff_g + (long long)l * D_MODEL, ln_ff_b + (long long)l * D_MODEL);
  }

  hipMemcpyAsync(d_out, h, TD * sizeof(float), hipMemcpyDeviceToDevice, stream);
}